// HybridQuantumGNN_35081292873758
// MI455X (gfx1250) — compile-verified
//
#include <hip/hip_runtime.h>

// ---------------------------------------------------------------------------
// 14-qubit statevector simulator, one workgroup (256 thr / 8 wave32) per batch
// element; statevector (16384 x complex64 = 128 KB) in LDS.
//
// Gate algebra baked in at compile time:
//  * embedding RX + layer-0 RX fused (RX(a)RX(b)=RX(a+b)) and applied
//    analytically as a product-state init (no gate sweeps).
//  * CNOT rings are GF(2)-linear basis permutations P -> never move data;
//    layer-1 RX on logical wire q pairs physical indices by mask P^{-1}(e_q),
//    measurement signs use rows of P^2.  RX is symmetric, so pair orientation
//    is irrelevant.
//  * layer-1 gates grouped 4+4+4+2.  Each 4-group acts on 16-element XOR
//    orbits as a Kronecker 16x16 complex unitary U[a][b] = f(a^b)  ->
//    D = U x S done with v_wmma_f32_16x16x4_f32 (4 chained K-chunks,
//    4 real matmuls per complex matmul).  Final 2-group (4-dim orbits) is
//    applied in registers and fused with the <Z_w> measurement (no store).
// ---------------------------------------------------------------------------

typedef float v2f __attribute__((ext_vector_type(2)));
typedef float v8f __attribute__((ext_vector_type(8)));

#define NQ      14
#define DIM     16384
#define THREADS 256

// --- compile-time GF(2) algebra for the CNOT ring -------------------------
constexpr unsigned ring_fwd(unsigned b) {
    for (int w = 0; w < NQ; ++w) b ^= ((b >> w) & 1u) << ((w + 1) % NQ);
    return b;
}
constexpr unsigned ring_inv(unsigned b) {
    for (int w = NQ - 1; w >= 0; --w) b ^= ((b >> w) & 1u) << ((w + 1) % NQ);
    return b;
}
constexpr int cparity(unsigned v) { int p = 0; while (v) { p ^= (int)(v & 1u); v >>= 1; } return p; }

struct Group4 { unsigned piv[4]; unsigned mix[16]; };
struct Group2 { unsigned piv[2]; unsigned mix[4]; };

constexpr Group4 make_group4(int q0) {
    unsigned m[4] = { ring_inv(1u << q0), ring_inv(1u << (q0 + 1)),
                      ring_inv(1u << (q0 + 2)), ring_inv(1u << (q0 + 3)) };
    unsigned e[4] = { m[0], m[1], m[2], m[3] };
    unsigned piv[4] = {};
    for (int k = 0; k < 4; ++k) {                       // Gaussian elimination
        unsigned p = 0; while (!((e[k] >> p) & 1u)) ++p;
        piv[k] = p;
        for (int j = 0; j < 4; ++j) if (j != k && ((e[j] >> p) & 1u)) e[j] ^= e[k];
    }
    Group4 t{};
    for (int a = 0; a < 4; ++a) t.piv[a] = piv[a];
    for (int a = 0; a < 4; ++a)                          // sort pivots ascending
        for (int b = a + 1; b < 4; ++b)
            if (t.piv[b] < t.piv[a]) { unsigned tmp = t.piv[a]; t.piv[a] = t.piv[b]; t.piv[b] = tmp; }
    for (unsigned c = 0; c < 16; ++c) {
        unsigned v = 0;
        for (int k = 0; k < 4; ++k) if ((c >> k) & 1u) v ^= m[k];
        t.mix[c] = v;
    }
    return t;
}

constexpr Group2 make_group2(int q0) {
    unsigned m[2] = { ring_inv(1u << q0), ring_inv(1u << (q0 + 1)) };
    unsigned e[2] = { m[0], m[1] };
    unsigned piv[2] = {};
    for (int k = 0; k < 2; ++k) {
        unsigned p = 0; while (!((e[k] >> p) & 1u)) ++p;
        piv[k] = p;
        for (int j = 0; j < 2; ++j) if (j != k && ((e[j] >> p) & 1u)) e[j] ^= e[k];
    }
    Group2 t{};
    t.piv[0] = piv[0] < piv[1] ? piv[0] : piv[1];
    t.piv[1] = piv[0] < piv[1] ? piv[1] : piv[0];
    t.mix[0] = 0; t.mix[1] = m[0]; t.mix[2] = m[1]; t.mix[3] = m[0] ^ m[1];
    return t;
}

struct AllTab {
    unsigned rowP2[NQ];   // parity(i & rowP2[w]) == bit_w(P(P(i)))
    Group4   g4[3];
    Group2   gd;
    unsigned selD[NQ];    // {g1,g2} sign-pattern selector for final group
};

constexpr AllTab make_all() {
    AllTab t{};
    for (int w = 0; w < NQ; ++w) {
        unsigned r2 = 0;
        for (int k = 0; k < NQ; ++k)
            if ((ring_fwd(ring_fwd(1u << k)) >> w) & 1u) r2 |= 1u << k;
        t.rowP2[w] = r2;
    }
    t.g4[0] = make_group4(0);
    t.g4[1] = make_group4(4);
    t.g4[2] = make_group4(8);
    t.gd    = make_group2(12);
    for (int w = 0; w < NQ; ++w)
        t.selD[w] = (unsigned)cparity(t.gd.mix[1] & t.rowP2[w])
                  | ((unsigned)cparity(t.gd.mix[2] & t.rowP2[w]) << 1);
    return t;
}

constexpr AllTab TB = make_all();

__device__ __forceinline__ unsigned ins0(unsigned t, unsigned p) {
    return ((t >> p) << (p + 1)) | (t & ((1u << p) - 1u));
}

// symmetric RX pair update: n0 = c*a + s*J(b), n1 = c*b + s*J(a), J(x,y)=(y,-x)
__device__ __forceinline__ void rx_pair(float2& a, float2& b, float cs, float sn) {
    float2 na, nb;
    na.x = fmaf(cs, a.x,  sn * b.y);
    na.y = fmaf(cs, a.y, -sn * b.x);
    nb.x = fmaf(cs, b.x,  sn * a.y);
    nb.y = fmaf(cs, b.y, -sn * a.x);
    a = na; b = nb;
}

__global__ __launch_bounds__(THREADS)
void qsim14_wmma_kernel(const float* __restrict__ x,    // (BATCH, 14)
                        const float* __restrict__ wts,  // (2, 14)
                        float* __restrict__ out)        // (BATCH, 14)
{
    __shared__ float2 sv[DIM];          // 128 KB statevector
    __shared__ float  red[8][NQ];

    const int b    = blockIdx.x;
    const int tid  = threadIdx.x;
    const int lane = tid & 31;
    const int wave = tid >> 5;
    const int hf   = lane >> 4;         // lane half (wave32)

    __builtin_prefetch(x + b * NQ, 0, 0);   // gfx1250 global_prefetch_b8
    __builtin_prefetch(wts, 0, 0);

    // ---- angles: fused (embedding + layer0) and layer1
    float c0[NQ], s0[NQ], c1[NQ], s1[NQ];
#pragma unroll
    for (int q = 0; q < NQ; ++q) {
        sincosf((x[b * NQ + q] + wts[q]) * 0.5f, &s0[q], &c0[q]);
        sincosf(wts[NQ + q] * 0.5f, &s1[q], &c1[q]);
    }

    // ---- analytic product-state init: amp(i) = (-i)^popc(i) * prod(cos/sin)
    for (int k = 0; k < DIM / THREADS; ++k) {
        int i = tid + k * THREADS;
        float pr = 1.0f;
#pragma unroll
        for (int q = 0; q < NQ; ++q) pr *= ((i >> q) & 1) ? s0[q] : c0[q];
        int pc = __popc((unsigned)i) & 3;
        float2 a;
        a.x = (pc == 0) ? pr : ((pc == 2) ? -pr : 0.0f);
        a.y = (pc == 1) ? -pr : ((pc == 3) ? pr : 0.0f);
        sv[i] = a;
    }
    __syncthreads();

    // ---- three 4-gate groups as 16x16 complex matmuls on 16-wide orbit chunks
#pragma unroll
    for (int g = 0; g < 3; ++g) {
        const int q0 = 4 * g;

        // Build A operands (per WMMA layout: VGPR j, lane -> M=lane&15, K=j+2*hf)
        // U[a][b] = f(a^b), f(d) = prod(cos/sin) * (-i)^popc(d)
        v2f Ar[4], Ai[4], Ani[4];
#pragma unroll
        for (int kc = 0; kc < 4; ++kc) {
#pragma unroll
            for (int j = 0; j < 2; ++j) {
                unsigned kcol = (unsigned)(kc * 4 + j + 2 * hf);
                unsigned d = ((unsigned)lane & 15u) ^ kcol;
                float mag = 1.0f;
#pragma unroll
                for (int t = 0; t < 4; ++t)
                    mag *= ((d >> t) & 1u) ? s1[q0 + t] : c1[q0 + t];
                int pc = __popc(d) & 3;
                float fr = (pc == 0) ? mag : ((pc == 2) ? -mag : 0.0f);
                float fi = (pc == 1) ? -mag : ((pc == 3) ? mag : 0.0f);
                Ar[kc][j] = fr; Ai[kc][j] = fi; Ani[kc][j] = -fi;
            }
        }

        // Each wave owns 8 chunks of 16 orbits (whole orbits -> no intra-group races)
        for (int ci = 0; ci < 8; ++ci) {
            int orbit = (wave * 8 + ci) * 16 + (lane & 15);   // 0..1023
            unsigned rep = (unsigned)orbit;
            rep = ins0(rep, TB.g4[g].piv[0]); rep = ins0(rep, TB.g4[g].piv[1]);
            rep = ins0(rep, TB.g4[g].piv[2]); rep = ins0(rep, TB.g4[g].piv[3]);

            // Load B (K = combo b, N = orbit): VGPR j, lane -> K=j+2*hf, N=lane&15
            v2f Br[4], Bi[4];
#pragma unroll
            for (int kc = 0; kc < 4; ++kc) {
#pragma unroll
                for (int j = 0; j < 2; ++j) {
                    unsigned mx = hf ? TB.g4[g].mix[kc * 4 + j + 2]
                                     : TB.g4[g].mix[kc * 4 + j];      // const-folded
                    float2 v = sv[rep ^ mx];
                    Br[kc][j] = v.x; Bi[kc][j] = v.y;
                }
            }

            // out_r = Ur*Sr + (-Ui)*Si ; out_i = Ur*Si + Ui*Sr
            v8f dr = {}; v8f di = {};
#pragma unroll
            for (int kc = 0; kc < 4; ++kc)
                dr = __builtin_amdgcn_wmma_f32_16x16x4_f32(false, Ar[kc],  false, Br[kc], (short)0, dr, false, false);
#pragma unroll
            for (int kc = 0; kc < 4; ++kc)
                dr = __builtin_amdgcn_wmma_f32_16x16x4_f32(false, Ani[kc], false, Bi[kc], (short)0, dr, false, false);
#pragma unroll
            for (int kc = 0; kc < 4; ++kc)
                di = __builtin_amdgcn_wmma_f32_16x16x4_f32(false, Ar[kc],  false, Bi[kc], (short)0, di, false, false);
#pragma unroll
            for (int kc = 0; kc < 4; ++kc)
                di = __builtin_amdgcn_wmma_f32_16x16x4_f32(false, Ai[kc],  false, Br[kc], (short)0, di, false, false);

            // Store D (M = output combo a = r + 8*hf, N = orbit)
#pragma unroll
            for (int r = 0; r < 8; ++r) {
                unsigned mx = hf ? TB.g4[g].mix[r + 8] : TB.g4[g].mix[r];
                float2 o; o.x = dr[r]; o.y = di[r];
                sv[rep ^ mx] = o;
            }
        }
        __syncthreads();
    }

    // ---- final 2-gate group (4-dim orbits) fused with measurement; no store
    float ev[NQ];
#pragma unroll
    for (int q = 0; q < NQ; ++q) ev[q] = 0.0f;

    for (int oi = 0; oi < 16; ++oi) {
        unsigned t = (unsigned)(tid + oi * THREADS);            // 0..4095
        unsigned rep = ins0(ins0(t, TB.gd.piv[0]), TB.gd.piv[1]);
        float2 v0 = sv[rep];
        float2 v1 = sv[rep ^ TB.gd.mix[1]];
        float2 v2 = sv[rep ^ TB.gd.mix[2]];
        float2 v3 = sv[rep ^ TB.gd.mix[3]];
        rx_pair(v0, v1, c1[12], s1[12]);   // gate 12: combo bit 0
        rx_pair(v2, v3, c1[12], s1[12]);
        rx_pair(v0, v2, c1[13], s1[13]);   // gate 13: combo bit 1
        rx_pair(v1, v3, c1[13], s1[13]);

        float p0 = fmaf(v0.x, v0.x, v0.y * v0.y);
        float p1 = fmaf(v1.x, v1.x, v1.y * v1.y);
        float p2 = fmaf(v2.x, v2.x, v2.y * v2.y);
        float p3 = fmaf(v3.x, v3.x, v3.y * v3.y);
        float S00 = (p0 + p1) + (p2 + p3);
        float S10 = (p0 - p1) + (p2 - p3);
        float S01 = (p0 + p1) - (p2 + p3);
        float S11 = (p0 - p1) - (p2 - p3);
#pragma unroll
        for (int w = 0; w < NQ; ++w) {
            float s = (TB.selD[w] == 0u) ? S00 :
                      (TB.selD[w] == 1u) ? S10 :
                      (TB.selD[w] == 2u) ? S01 : S11;           // const-folded
            bool neg = (__popc(rep & TB.rowP2[w]) & 1) != 0;
            ev[w] += neg ? -s : s;
        }
    }

    // ---- wave32 shuffle reduce, cross-wave LDS reduce, softplus, store
#pragma unroll
    for (int q = 0; q < NQ; ++q) {
        float v = ev[q];
        for (int off = 16; off > 0; off >>= 1) v += __shfl_down(v, off, 32);
        ev[q] = v;
    }
    if ((tid & 31) == 0) {
#pragma unroll
        for (int q = 0; q < NQ; ++q) red[wave][q] = ev[q];
    }
    __syncthreads();

    if (tid < NQ) {
        float s = 0.0f;
#pragma unroll
        for (int wv = 0; wv < 8; ++wv) s += red[wv][tid];
        float sp = fmaxf(s, 0.0f) + log1pf(expf(-fabsf(s)));
        out[b * NQ + tid] = sp;
    }
}

extern "C" void kernel_launch(void* const* d_in, const int* in_sizes, int n_in,
                              void* d_out, int out_size, void* d_ws, size_t ws_size,
                              hipStream_t stream) {
    const float* x   = (const float*)d_in[0];   // (1024, 14) fp32
    const float* wts = (const float*)d_in[1];   // (2, 14)    fp32
    float* out = (float*)d_out;                 // (1024, 14) fp32

    const int batch = in_sizes[0] / NQ;         // 1024
    qsim14_wmma_kernel<<<batch, THREADS, 0, stream>>>(x, wts, out);
}